// MPSPeriodic_59382217835186
// MI455X (gfx1250) — compile-verified
//
#include <hip/hip_runtime.h>
#include <math.h>

typedef float v2f __attribute__((ext_vector_type(2)));
typedef float v4f __attribute__((ext_vector_type(4)));
typedef float v8f __attribute__((ext_vector_type(8)));

#define B_CONST 1024
#define L_CONST 1024
#define PHYS    2
#define D       16
#define DD      (D * D)
#define WPB     8            // wave32s per block
#define NSEG    8            // segments per chain (workspace path)
#define NMAT    (L_CONST * PHYS)

// One chain step: carry_new = A * carry.  A (16x16) held as 4 K-group chunks
// a[g] covering columns {2g, 2g+1, 2g+8, 2g+9}; the matching 4x16 B operand
// for group g is exactly C/D-layout VGPRs {2g, 2g+1} of carry.
__device__ __forceinline__ v8f chain_step(const v2f a[4], v8f carry) {
    v8f acc = {};
#pragma unroll
    for (int g = 0; g < 4; ++g) {
        v2f bg;
        bg.x = carry[2 * g];
        bg.y = carry[2 * g + 1];
        // 8 args: (neg_a, A, neg_b, B, c_mod, C, reuse_a, reuse_b)
        acc = __builtin_amdgcn_wmma_f32_16x16x4_f32(
            false, a[g], false, bg, (short)0, acc, false, false);
    }
    return acc;
}

// Untransposed gather (p = M + rowbase*D + col): a[g].x = M[2g + 8hi, col]
__device__ __forceinline__ void load_a_T(v2f a[4], const float* __restrict__ p) {
#pragma unroll
    for (int g = 0; g < 4; ++g) {
        a[g].x = p[(2 * g) * D];
        a[g].y = p[(2 * g + 1) * D];
    }
}

// Pre-transposed gather: lane's 8 A values contiguous -> 2x b128 loads.
__device__ __forceinline__ void load_a_tk(v2f a[4], const float* __restrict__ tp) {
    v4f t0 = *(const v4f*)tp;
    v4f t1 = *(const v4f*)(tp + 4);
    a[0].x = t0.x; a[0].y = t0.y;
    a[1].x = t0.z; a[1].y = t0.w;
    a[2].x = t1.x; a[2].y = t1.y;
    a[3].x = t1.z; a[3].y = t1.w;
}

__device__ __forceinline__ float trace_cd(v8f carry, int rowbase, int col) {
    float t = 0.0f;
#pragma unroll
    for (int v = 0; v < 8; ++v)
        t += ((v + rowbase) == col) ? carry[v] : 0.0f;
#pragma unroll
    for (int off = 16; off > 0; off >>= 1)
        t += __shfl_xor(t, off, 32);
    return t;
}

// One-time (per launch) relayout of kernel[l,p] into A-operand lane order:
// tk[mat*256 + lane*8 + r] = M[r + 8*(lane>>4), lane&15],  r = 0..7
__global__ __launch_bounds__(WPB * 32)
void mps_transpose_kernel(const float* __restrict__ kern, float* __restrict__ tk) {
    const int lane = threadIdx.x & 31;
    const int wave = threadIdx.x >> 5;
    const int m    = blockIdx.x * WPB + wave;        // matrix index 0..NMAT-1
    const int hi = lane >> 4, col = lane & 15, rowbase = hi << 3;

    const float* src = kern + (size_t)m * DD + rowbase * D + col;
    float*       dst = tk   + (size_t)m * DD + lane * 8;
    float t[8];
#pragma unroll
    for (int r = 0; r < 8; ++r) t[r] = src[r * D];
#pragma unroll
    for (int r = 0; r < 8; ++r) dst[r] = t[r];
}

// One wave computes the ordered product of SEG transposed matrices.
// TK    : matrices come pre-transposed from workspace (2x b128 per step).
// FINAL : SEG == L, dst = out, write log(trace) directly; else store partial.
template <int SEG, bool FINAL, bool TK>
__global__ __launch_bounds__(WPB * 32)
void mps_segment_kernel(const int* __restrict__ x,
                        const float* __restrict__ kern,
                        float* __restrict__ dst) {
    constexpr int S    = L_CONST / SEG;   // segments per chain
    constexpr int NREG = SEG / 32;        // 32-step chunks

    const int lane = threadIdx.x & 31;
    const int wave = threadIdx.x >> 5;
    const int gid  = blockIdx.x * WPB + wave;
    const int b    = gid / S;
    const int s    = gid % S;

    const int hi       = lane >> 4;
    const int col      = lane & 15;
    const int rowbase  = hi << 3;
    const int lane_off = TK ? (lane * 8) : (rowbase * D + col);

    const size_t xbase = (size_t)b * L_CONST + (size_t)s * SEG;
    const int    lbase = s * SEG;

    // carry = Identity in C/D layout (vgpr v -> M = v + 8*hi, N = col)
    v8f carry;
#pragma unroll
    for (int v = 0; v < 8; ++v)
        carry[v] = ((v + rowbase) == col) ? 1.0f : 0.0f;

    // x selectors in VGPRs: one coalesced b32 load per 32-step chunk,
    // extracted per step with v_readlane at immediate lane indices.
    int xc = x[xbase + lane];                        // chunk 0

    v2f a_cur[4], a_nxt[4];
    {
        int xv = __builtin_amdgcn_readlane(xc, 0);
        const float* p = kern + ((size_t)lbase * PHYS + xv) * DD + lane_off;
        if (TK) load_a_tk(a_cur, p); else load_a_T(a_cur, p);
    }

    for (int r = 0; r < NREG - 1; ++r) {
        int xn = x[xbase + 32 * (r + 1) + lane];     // issued a full chunk early
#pragma unroll
        for (int j = 0; j < 32; ++j) {
            const int xv = (j < 31) ? __builtin_amdgcn_readlane(xc, j + 1)
                                    : __builtin_amdgcn_readlane(xn, 0);
            const int l1 = lbase + r * 32 + j + 1;   // next step's l
            const float* p = kern + ((size_t)l1 * PHYS + xv) * DD + lane_off;
            if (TK) load_a_tk(a_nxt, p); else load_a_T(a_nxt, p);
            carry = chain_step(a_cur, carry);
#pragma unroll
            for (int g = 0; g < 4; ++g) a_cur[g] = a_nxt[g];
        }
        xc = xn;
    }
    {   // last chunk: 31 pipelined steps + drain
        const int rb = NREG - 1;
#pragma unroll
        for (int j = 0; j < 31; ++j) {
            const int xv = __builtin_amdgcn_readlane(xc, j + 1);
            const int l1 = lbase + rb * 32 + j + 1;
            const float* p = kern + ((size_t)l1 * PHYS + xv) * DD + lane_off;
            if (TK) load_a_tk(a_nxt, p); else load_a_T(a_nxt, p);
            carry = chain_step(a_cur, carry);
#pragma unroll
            for (int g = 0; g < 4; ++g) a_cur[g] = a_nxt[g];
        }
        carry = chain_step(a_cur, carry);
    }

    if constexpr (FINAL) {
        float t = trace_cd(carry, rowbase, col);
        if (lane == 0) dst[b] = logf(t);
    } else {
        // store partial product row-major: W[m*16 + n] = carry_cd(m, n)
        float* Wd = dst + ((size_t)b * S + s) * DD;
#pragma unroll
        for (int v = 0; v < 8; ++v)
            Wd[(v + rowbase) * D + col] = carry[v];
    }
}

// Ordered product of the NSEG per-segment matrices (ascending s gives
// seg_{S-1} * ... * seg_0 = final^T), then log(trace).
__global__ __launch_bounds__(WPB * 32)
void mps_combine_kernel(const float* __restrict__ ws, float* __restrict__ out) {
    const int lane = threadIdx.x & 31;
    const int wave = threadIdx.x >> 5;
    const int b    = blockIdx.x * WPB + wave;

    const int hi      = lane >> 4;
    const int col     = lane & 15;      // serves as A-row index m
    const int rowbase = hi << 3;

    v8f Q;
#pragma unroll
    for (int v = 0; v < 8; ++v)
        Q[v] = ((v + rowbase) == col) ? 1.0f : 0.0f;

    // A[m, 8hi + c] for c=0..7 are 8 contiguous floats per lane.
    const float* Wb = ws + (size_t)b * NSEG * DD + col * D + 8 * hi;
#pragma unroll
    for (int s = 0; s < NSEG; ++s) {
        const float* p = Wb + s * DD;
        v2f a[4];
#pragma unroll
        for (int g = 0; g < 4; ++g) {
            a[g].x = p[2 * g];
            a[g].y = p[2 * g + 1];
        }
        Q = chain_step(a, Q);
    }

    float t = trace_cd(Q, rowbase, col);
    if (lane == 0) out[b] = logf(t);
}

extern "C" void kernel_launch(void* const* d_in, const int* in_sizes, int n_in,
                              void* d_out, int out_size, void* d_ws, size_t ws_size,
                              hipStream_t stream) {
    (void)in_sizes; (void)n_in; (void)out_size;
    const int*   x    = (const int*)d_in[0];
    const float* kern = (const float*)d_in[1];
    float*       out  = (float*)d_out;

    const size_t tk_elems = (size_t)NMAT * DD;                 // 512K floats (2 MB)
    const size_t pp_elems = (size_t)B_CONST * NSEG * DD;       // 2M floats (8 MB)
    const size_t ws_need  = (tk_elems + pp_elems) * sizeof(float); // 10 MB

    if (ws_size >= ws_need && d_ws != nullptr) {
        float* tk = (float*)d_ws;
        float* pp = tk + tk_elems;
        mps_transpose_kernel
            <<<dim3(NMAT / WPB), dim3(WPB * 32), 0, stream>>>(kern, tk);
        // 8192 waves, 2x b128 A-tile loads per step.
        mps_segment_kernel<L_CONST / NSEG, false, true>
            <<<dim3(B_CONST * NSEG / WPB), dim3(WPB * 32), 0, stream>>>(x, tk, pp);
        mps_combine_kernel
            <<<dim3(B_CONST / WPB), dim3(WPB * 32), 0, stream>>>(pp, out);
    } else {
        // Fallback: one wave per full chain, untransposed gathers, finalize in-place.
        mps_segment_kernel<L_CONST, true, false>
            <<<dim3(B_CONST / WPB), dim3(WPB * 32), 0, stream>>>(x, kern, out);
    }
}